// Discriminator_54365696033560
// MI455X (gfx1250) — compile-verified
//
#include <hip/hip_runtime.h>
#include <math.h>

typedef __attribute__((ext_vector_type(2))) float v2f;
typedef __attribute__((ext_vector_type(8))) float v8f;

#define FEAT  512
#define HID   1024
#define KD    15
#define NKER  5
#define DCAT  1029
#define LDACT 1056          // DCAT padded to multiple of 32 (K of GEMM1)
#define LRELU 0.3f
#define LNEPS 1e-3f

// ---------------- WMMA fp32 GEMM: C = A(MxK) @ B(KxN) + bias ----------------
#define BM 128
#define BN 128
#define BK 32
#define SA 36     // A tile LDS stride (mult of 4 for float4 stores; 36r mod 64 distinct over 16 rows)
#define SBP 288   // interleaved B tile stride per K-pair row; 288 % 64 == 32 -> half-waves on disjoint banks

__device__ __forceinline__ void store_acc(float* __restrict__ C, int ldc, int row0,
                                          int col, const float* __restrict__ bias, v8f acc) {
    float bv = bias[col];
#pragma unroll
    for (int r = 0; r < 8; ++r)
        C[(size_t)(row0 + r) * ldc + col] = acc[r] + bv;
}

__global__ __launch_bounds__(256) void gemm_wmma_f32(
    const float* __restrict__ A, const float* __restrict__ B,
    const float* __restrict__ bias, float* __restrict__ C,
    int K, int lda, int ldb, int ldc)
{
    __shared__ float As[BM * SA];            // 18.4 KB
    __shared__ float Bs[(BK / 2) * SBP];     // 18.4 KB, K-pair interleaved: B[k][c] -> [(k>>1)*SBP + 2c + (k&1)]

    const int tid  = threadIdx.x;
    const int lane = tid & 31;
    const int wv   = tid >> 5;      // 8 waves
    const int hl   = lane >> 4;     // half-wave: 0 or 1
    const int l16  = lane & 15;

    const int rowBase = blockIdx.y * BM;
    const int colBase = blockIdx.x * BN;
    const int wm = (wv & 3) * 32;   // wave's M offset (2 sub-tiles of 16)
    const int wn = (wv >> 2) * 64;  // wave's N offset (4 sub-tiles of 16)

    v8f acc[2][4] = {};

    for (int k0 = 0; k0 < K; k0 += BK) {
        __syncthreads();
        // A tile: 128x32 floats = 1024 float4, 4 per thread (row-major, stride SA)
#pragma unroll
        for (int p = 0; p < 4; ++p) {
            int idx = tid + p * 256;
            int r = idx >> 3;
            int c = (idx & 7) << 2;
            float4 v = *reinterpret_cast<const float4*>(A + (size_t)(rowBase + r) * lda + (k0 + c));
            *reinterpret_cast<float4*>(&As[r * SA + c]) = v;
        }
        // B tile: 32x128 floats, K-pair interleaved so a fragment (K=kb,kb+1 at one col) is one b64
#pragma unroll
        for (int p = 0; p < 4; ++p) {
            int idx = tid + p * 256;
            int r = idx >> 5;
            int c = (idx & 31) << 2;
            float4 v = *reinterpret_cast<const float4*>(B + (size_t)(k0 + r) * ldb + (colBase + c));
            int base = (r >> 1) * SBP + (r & 1);
            Bs[base + (c + 0) * 2] = v.x;
            Bs[base + (c + 1) * 2] = v.y;
            Bs[base + (c + 2) * 2] = v.z;
            Bs[base + (c + 3) * 2] = v.w;
        }
        // prefetch next K-chunk into cache (global_prefetch path) while we compute this one
        if (k0 + BK < K) {
#pragma unroll
            for (int p = 0; p < 4; ++p) {
                int idx = tid + p * 256;
                __builtin_prefetch(A + (size_t)(rowBase + (idx >> 3)) * lda + (k0 + BK) + ((idx & 7) << 2), 0, 3);
                __builtin_prefetch(B + (size_t)(k0 + BK + (idx >> 5)) * ldb + colBase + ((idx & 31) << 2), 0, 3);
            }
        }
        __syncthreads();

#pragma unroll
        for (int kk = 0; kk < BK; kk += 4) {
            const int kb = kk + (hl << 1);   // lanes 0-15: K=kk,kk+1 ; lanes 16-31: K=kk+2,kk+3
            v2f a0 = *reinterpret_cast<const v2f*>(&As[(wm      + l16) * SA + kb]);
            v2f a1 = *reinterpret_cast<const v2f*>(&As[(wm + 16 + l16) * SA + kb]);
            const int brow = (kb >> 1) * SBP;
            v2f b0 = *reinterpret_cast<const v2f*>(&Bs[brow + (wn      + l16) * 2]);
            v2f b1 = *reinterpret_cast<const v2f*>(&Bs[brow + (wn + 16 + l16) * 2]);
            v2f b2 = *reinterpret_cast<const v2f*>(&Bs[brow + (wn + 32 + l16) * 2]);
            v2f b3 = *reinterpret_cast<const v2f*>(&Bs[brow + (wn + 48 + l16) * 2]);
            acc[0][0] = __builtin_amdgcn_wmma_f32_16x16x4_f32(false, a0, false, b0, (short)0, acc[0][0], false, false);
            acc[0][1] = __builtin_amdgcn_wmma_f32_16x16x4_f32(false, a0, false, b1, (short)0, acc[0][1], false, false);
            acc[0][2] = __builtin_amdgcn_wmma_f32_16x16x4_f32(false, a0, false, b2, (short)0, acc[0][2], false, false);
            acc[0][3] = __builtin_amdgcn_wmma_f32_16x16x4_f32(false, a0, false, b3, (short)0, acc[0][3], false, false);
            acc[1][0] = __builtin_amdgcn_wmma_f32_16x16x4_f32(false, a1, false, b0, (short)0, acc[1][0], false, false);
            acc[1][1] = __builtin_amdgcn_wmma_f32_16x16x4_f32(false, a1, false, b1, (short)0, acc[1][1], false, false);
            acc[1][2] = __builtin_amdgcn_wmma_f32_16x16x4_f32(false, a1, false, b2, (short)0, acc[1][2], false, false);
            acc[1][3] = __builtin_amdgcn_wmma_f32_16x16x4_f32(false, a1, false, b3, (short)0, acc[1][3], false, false);
        }
    }

#pragma unroll
    for (int mi = 0; mi < 2; ++mi) {
        const int row0 = rowBase + wm + mi * 16 + hl * 8;   // C/D: VGPR r -> M=r (lanes<16) / M=r+8
        store_acc(C, ldc, row0, colBase + wn +  0 + l16, bias, acc[mi][0]);
        store_acc(C, ldc, row0, colBase + wn + 16 + l16, bias, acc[mi][1]);
        store_acc(C, ldc, row0, colBase + wn + 32 + l16, bias, acc[mi][2]);
        store_acc(C, ldc, row0, colBase + wn + 48 + l16, bias, acc[mi][3]);
    }
}

// ---------------- pack w1_a (1029x1024) into zero-padded (1056x1024) ----------------
__global__ void padw_kernel(const float* __restrict__ W, float* __restrict__ Wp) {
    int idx = blockIdx.x * blockDim.x + threadIdx.x;
    if (idx >= LDACT * HID) return;
    int row = idx / HID;
    Wp[idx] = (row < DCAT) ? W[idx] : 0.0f;
}

// ---------------- small dense: T(Mx16, col15=0) = H(MxHID) @ W(HIDx15) + b ----------------
__global__ __launch_bounds__(256) void dense15_kernel(
    const float* __restrict__ H, const float* __restrict__ W,
    const float* __restrict__ b, float* __restrict__ T, int M)
{
    int idx = blockIdx.x * blockDim.x + threadIdx.x;
    int i = idx >> 4;
    int n = idx & 15;
    if (i >= M) return;
    float s = 0.0f;
    if (n < KD) {
        const float* hrow = H + (size_t)i * HID;
        for (int k = 0; k < HID; ++k)
            s = fmaf(hrow[k], W[k * KD + n], s);
        s += b[n];
    }
    T[(size_t)i * 16 + n] = (n < KD) ? s : 0.0f;
}

// ---------------- zero the diversity accumulator ----------------
__global__ void zero_kernel(float* __restrict__ p, int n) {
    int idx = blockIdx.x * blockDim.x + threadIdx.x;
    if (idx < n) p[idx] = 0.0f;
}

// ---- minibatch diversity, j-split for parallelism: Dv[i][k] += sum_{j in split} exp(-L1) ----
#define JT 256
#define JSPLIT 16
__global__ __launch_bounds__(256) void diversity_kernel(
    const float* __restrict__ T, float* __restrict__ Dv, int M)
{
    __shared__ float sj[JT * 16];
    const int i = blockIdx.x * blockDim.x + threadIdx.x;
    const int jlen = M / JSPLIT;
    const int jbeg = blockIdx.y * jlen;

    float ti[KD];
#pragma unroll
    for (int c = 0; c < KD; ++c) ti[c] = T[(size_t)i * 16 + c];

    float acc[NKER] = {0.f, 0.f, 0.f, 0.f, 0.f};

    for (int j0 = jbeg; j0 < jbeg + jlen; j0 += JT) {
        __syncthreads();
#pragma unroll
        for (int p = 0; p < 4; ++p) {   // JT*16 floats = 1024 float4, 4 per thread
            int idx = threadIdx.x + p * 256;
            *reinterpret_cast<float4*>(&sj[idx * 4]) =
                *reinterpret_cast<const float4*>(T + (size_t)j0 * 16 + idx * 4);
        }
        __syncthreads();

        for (int jj = 0; jj < JT; ++jj) {
            const float* tj = &sj[jj * 16];
#pragma unroll
            for (int k = 0; k < NKER; ++k) {
                float s = fabsf(ti[3 * k + 0] - tj[3 * k + 0])
                        + fabsf(ti[3 * k + 1] - tj[3 * k + 1])
                        + fabsf(ti[3 * k + 2] - tj[3 * k + 2]);
                acc[k] += __expf(-s);
            }
        }
    }
#pragma unroll
    for (int k = 0; k < NKER; ++k)
        atomicAdd(&Dv[(size_t)i * 8 + k], acc[k]);
}

// ---------------- concat(h, div) -> LayerNorm(center-only) -> LeakyReLU, padded store ----------------
__global__ __launch_bounds__(256) void concat_ln_lrelu_kernel(
    const float* __restrict__ H, const float* __restrict__ Dv,
    const float* __restrict__ beta, float* __restrict__ Act)
{
    const int row = blockIdx.x;
    const int tid = threadIdx.x;
    __shared__ float rs[256];
    __shared__ float rq[256];

    float s = 0.f, q = 0.f;
    for (int c = tid; c < DCAT; c += 256) {
        float v = (c < HID) ? H[(size_t)row * HID + c] : Dv[(size_t)row * 8 + (c - HID)];
        s += v;
        q += v * v;
    }
    rs[tid] = s; rq[tid] = q;
    __syncthreads();
    for (int st = 128; st > 0; st >>= 1) {
        if (tid < st) { rs[tid] += rs[tid + st]; rq[tid] += rq[tid + st]; }
        __syncthreads();
    }
    const float inv_n = 1.0f / (float)DCAT;
    float mu  = rs[0] * inv_n;
    float var = rq[0] * inv_n - mu * mu;
    float rinv = rsqrtf(var + LNEPS);

    for (int c = tid; c < DCAT; c += 256) {
        float v = (c < HID) ? H[(size_t)row * HID + c] : Dv[(size_t)row * 8 + (c - HID)];
        float y = (v - mu) * rinv + beta[c];
        y = (y >= 0.f) ? y : LRELU * y;
        Act[(size_t)row * LDACT + c] = y;
    }
    for (int c = DCAT + tid; c < LDACT; c += 256)
        Act[(size_t)row * LDACT + c] = 0.0f;   // zero K-padding for GEMM1
}

// ---------------- final head: out[i] = act[i,:] @ wf + bf (one wave per row) ----------------
__global__ __launch_bounds__(256) void final_head_kernel(
    const float* __restrict__ Act, const float* __restrict__ wf,
    const float* __restrict__ bf, float* __restrict__ out, int M)
{
    const int row  = (blockIdx.x * blockDim.x + threadIdx.x) >> 5;
    const int lane = threadIdx.x & 31;
    if (row >= M) return;
    float s = 0.f;
    for (int c = lane; c < DCAT; c += 32)
        s = fmaf(Act[(size_t)row * LDACT + c], wf[c], s);
#pragma unroll
    for (int off = 16; off > 0; off >>= 1)
        s += __shfl_down(s, off, 32);
    if (lane == 0) out[row] = s + bf[0];
}

// ---------------------------------------------------------------------------
extern "C" void kernel_launch(void* const* d_in, const int* in_sizes, int n_in,
                              void* d_out, int out_size, void* d_ws, size_t ws_size,
                              hipStream_t stream)
{
    const float* x    = (const float*)d_in[0];
    const float* w0a  = (const float*)d_in[1];
    const float* b0a  = (const float*)d_in[2];
    const float* w0b  = (const float*)d_in[3];
    const float* b0b  = (const float*)d_in[4];
    const float* bet0 = (const float*)d_in[5];
    const float* w1a  = (const float*)d_in[6];
    const float* b1a  = (const float*)d_in[7];
    const float* w1b  = (const float*)d_in[8];
    const float* b1b  = (const float*)d_in[9];
    const float* bet1 = (const float*)d_in[10];
    const float* wf   = (const float*)d_in[11];
    const float* bf   = (const float*)d_in[12];
    float* out = (float*)d_out;

    const int M = in_sizes[0] / FEAT;   // 4096

    float* ws    = (float*)d_ws;
    float* hbuf  = ws;                                  // M x HID   (reused both layers)
    float* tbuf  = hbuf  + (size_t)M * HID;             // M x 16
    float* dvbuf = tbuf  + (size_t)M * 16;              // M x 8
    float* actbf = dvbuf + (size_t)M * 8;               // M x LDACT (reused both layers)
    float* wpad  = actbf + (size_t)M * LDACT;           // LDACT x HID

    dim3 blk(256);
    dim3 gemm_grid(HID / BN, M / BM);
    dim3 div_grid(M / 256, JSPLIT);

    padw_kernel<<<dim3((LDACT * HID + 255) / 256), blk, 0, stream>>>(w1a, wpad);

    // ---- layer 0 ----
    gemm_wmma_f32<<<gemm_grid, blk, 0, stream>>>(x, w0a, b0a, hbuf, FEAT, FEAT, HID, HID);
    dense15_kernel<<<dim3((M * 16 + 255) / 256), blk, 0, stream>>>(hbuf, w0b, b0b, tbuf, M);
    zero_kernel<<<dim3((M * 8 + 255) / 256), blk, 0, stream>>>(dvbuf, M * 8);
    diversity_kernel<<<div_grid, blk, 0, stream>>>(tbuf, dvbuf, M);
    concat_ln_lrelu_kernel<<<dim3(M), blk, 0, stream>>>(hbuf, dvbuf, bet0, actbf);

    // ---- layer 1 ----
    gemm_wmma_f32<<<gemm_grid, blk, 0, stream>>>(actbf, wpad, b1a, hbuf, LDACT, LDACT, HID, HID);
    dense15_kernel<<<dim3((M * 16 + 255) / 256), blk, 0, stream>>>(hbuf, w1b, b1b, tbuf, M);
    zero_kernel<<<dim3((M * 8 + 255) / 256), blk, 0, stream>>>(dvbuf, M * 8);
    diversity_kernel<<<div_grid, blk, 0, stream>>>(tbuf, dvbuf, M);
    concat_ln_lrelu_kernel<<<dim3(M), blk, 0, stream>>>(hbuf, dvbuf, bet1, actbf);

    // ---- critic value ----
    final_head_kernel<<<dim3(M / 8), blk, 0, stream>>>(actbf, wf, bf, out, M);
}